// DeepwiseAtn_28836410425470
// MI455X (gfx1250) — compile-verified
//
#include <hip/hip_runtime.h>
#include <hip/hip_bf16.h>

// MI455X / gfx1250, wave32. Problem: B=16, C=256, H=W=64 (HW=4096), OUT=256.
//   scale[b,c] = mean(context[b,c,:,:])
//   out[b]     = w1 @ (diag(scale[b]) * x[b]) + b1     (per-batch GEMM 256x256x4096)
// Memory-bound (805 MB traffic -> ~35us floor @ 23.3 TB/s), so use exact fp32
// WMMA (V_WMMA_F32_16X16X4_F32) and fold scale into the 4MB weight tensor.

typedef __attribute__((ext_vector_type(2))) float v2f;
typedef __attribute__((ext_vector_type(8))) float v8f;

#define BATCH 16
#define CIN   256
#define COUT  256
#define HW    4096

// ---------------------------------------------------------------------------
// Kernel 1: scale[b,c] = mean over HW of context. One wave32 per (b,c) pair,
// 8 pairs per 256-thread block. float4 loads, shfl_xor tree reduce.
// ---------------------------------------------------------------------------
__global__ __launch_bounds__(256) void mean_kernel(const float* __restrict__ ctx,
                                                   float* __restrict__ scale) {
    int pair = blockIdx.x * 8 + (threadIdx.x >> 5);   // b*C + c
    int lane = threadIdx.x & 31;
    const float4* p = (const float4*)(ctx + (size_t)pair * HW);
    float s = 0.0f;
    #pragma unroll 4
    for (int i = lane; i < HW / 4; i += 32) {
        float4 v = p[i];
        s += (v.x + v.y) + (v.z + v.w);
    }
    #pragma unroll
    for (int off = 16; off > 0; off >>= 1)
        s += __shfl_xor(s, off, 32);
    if (lane == 0)
        scale[pair] = s * (1.0f / (float)HW);
}

// ---------------------------------------------------------------------------
// Kernel 2: w1s[b,o,c] = w1[o,c] * scale[b,c]   (4 MB, float4 elementwise)
// ---------------------------------------------------------------------------
__global__ __launch_bounds__(256) void scalew_kernel(const float* __restrict__ w1,
                                                     const float* __restrict__ scale,
                                                     float* __restrict__ w1s) {
    int idx4 = blockIdx.x * 256 + threadIdx.x;        // one float4 each
    int e    = idx4 * 4;                               // flat element index
    int c    = e % CIN;
    int o    = (e / CIN) % COUT;
    int b    = e / (CIN * COUT);
    float4 w = *(const float4*)(w1 + (size_t)o * CIN + c);
    float4 s = *(const float4*)(scale + (size_t)b * CIN + c);
    float4 r = make_float4(w.x * s.x, w.y * s.y, w.z * s.z, w.w * s.w);
    *(float4*)(w1s + ((size_t)b * COUT + o) * CIN + c) = r;
}

// ---------------------------------------------------------------------------
// Kernel 3: per-batch GEMM out[b] = w1s[b] (256x256) @ x[b] (256x4096) + b1
// using V_WMMA_F32_16X16X4_F32.
//   Each wave32 computes a 64(M) x 16(N) strip: 4 accumulators of v8f.
//   A-tile 16x4 f32 layout: lanes 0-15 = M rows, VGPR0/1 = K {khalf, khalf+1}
//     with khalf = (lane>=16 ? 2 : 0)  -> per-lane contiguous float2 load.
//   B-tile 4x16 f32 layout (mirror of A): lanes 0-15 = N cols, per-lane dwords
//     at K = khalf, khalf+1 (stride HW between K rows).
//   C/D 16x16 f32 layout: VGPR j holds row M = j + (lane>=16 ? 8 : 0).
// Bias is preloaded into the accumulators (D = A*B + C epilogue for free).
// ---------------------------------------------------------------------------
__global__ __launch_bounds__(256) void gemm_kernel(const float* __restrict__ x,
                                                   const float* __restrict__ w1s,
                                                   const float* __restrict__ b1,
                                                   float* __restrict__ out) {
    const int lane   = threadIdx.x & 31;
    const int wave   = threadIdx.x >> 5;
    const int b      = blockIdx.z;
    const int mBase  = blockIdx.y * 64;                    // 4 M-tiles of 16
    const int nBase  = (blockIdx.x * 8 + wave) * 16;       // 16 N columns

    const int lane16 = lane & 15;
    const int khalf  = (lane >> 4) << 1;                   // 0 or 2
    const int moff   = (lane >> 4) * 8;                    // C/D row offset

    // Per-lane base pointers.
    const float* Abase = w1s + ((size_t)b * COUT + mBase + lane16) * CIN + khalf;
    const float* Bbase = x + (size_t)b * CIN * HW + (size_t)khalf * HW + nBase + lane16;

    // Init accumulators with bias: acc[t][j] = b1[mBase + t*16 + moff + j]
    v8f acc[4];
    #pragma unroll
    for (int t = 0; t < 4; ++t) {
        #pragma unroll
        for (int j = 0; j < 8; ++j)
            acc[t][j] = b1[mBase + t * 16 + moff + j];
    }

    // K loop: 4 per step, 4 back-to-back WMMAs sharing one B tile.
    for (int k0 = 0; k0 < CIN; k0 += 4) {
        v2f bvec;
        bvec.x = Bbase[(size_t)k0 * HW];          // K = k0 + khalf
        bvec.y = Bbase[(size_t)(k0 + 1) * HW];    // K = k0 + khalf + 1
        #pragma unroll
        for (int t = 0; t < 4; ++t) {
            v2f avec = *(const v2f*)(Abase + (size_t)t * 16 * CIN + k0);
            acc[t] = __builtin_amdgcn_wmma_f32_16x16x4_f32(
                /*neg_a=*/false, avec, /*neg_b=*/false, bvec,
                /*c_mod=*/(short)0, acc[t], /*reuse_a=*/false, /*reuse_b=*/false);
        }
    }

    // Store: out[b, mRow, nBase + lane16], dword per VGPR row, coalesced
    // across lanes 0-15 / 16-31.
    #pragma unroll
    for (int t = 0; t < 4; ++t) {
        const int mRow = mBase + t * 16 + moff;
        float* o = out + ((size_t)b * COUT + mRow) * HW + nBase + lane16;
        #pragma unroll
        for (int j = 0; j < 8; ++j)
            o[(size_t)j * HW] = acc[t][j];
    }
}

// ---------------------------------------------------------------------------
// Launch
// ---------------------------------------------------------------------------
extern "C" void kernel_launch(void* const* d_in, const int* in_sizes, int n_in,
                              void* d_out, int out_size, void* d_ws, size_t ws_size,
                              hipStream_t stream) {
    const float* x   = (const float*)d_in[0];   // [B, C, H, W]
    const float* ctx = (const float*)d_in[1];   // [B, C, H, W]
    const float* w1  = (const float*)d_in[2];   // [OUT, C]
    const float* b1  = (const float*)d_in[3];   // [OUT]
    float* out = (float*)d_out;                 // [B, OUT, H, W]

    // Workspace layout: scale (B*C = 4096 floats, 16KB) then w1s (4MB).
    float* scale = (float*)d_ws;
    float* w1s   = scale + (size_t)BATCH * CIN;

    // 1) per-(b,c) mean: 4096 pairs, 8 waves/block -> 512 blocks.
    mean_kernel<<<dim3(BATCH * CIN / 8), dim3(256), 0, stream>>>(ctx, scale);

    // 2) scaled weights: B*OUT*C/4 float4 elements -> 1024 blocks.
    scalew_kernel<<<dim3(BATCH * COUT * CIN / 4 / 256), dim3(256), 0, stream>>>(w1, scale, w1s);

    // 3) WMMA GEMM: grid (HW/(16*8)=32, OUT/64=4, B=16), 8 waves/block.
    gemm_kernel<<<dim3(HW / 128, COUT / 64, BATCH), dim3(256), 0, stream>>>(x, w1s, b1, out);
}